// GCNConv_25202868093076
// MI455X (gfx1250) — compile-verified
//
#include <hip/hip_runtime.h>

#define F_IN  128
#define F_OUT 128

typedef __attribute__((ext_vector_type(2))) float v2f;
typedef __attribute__((ext_vector_type(8))) float v8f;

// ---------------------------------------------------------------------------
// Phase 0: zero the support accumulator (in d_out) and the degree array (d_ws)
// ---------------------------------------------------------------------------
__global__ void gcn_zero_kernel(float* __restrict__ support, float* __restrict__ deg,
                                int n_support, int n_deg) {
    int i = blockIdx.x * blockDim.x + threadIdx.x;
    int stride = gridDim.x * blockDim.x;
    float4* s4 = (float4*)support;
    int n4 = n_support >> 2;                       // n_support divisible by 4
    for (int j = i; j < n4; j += stride) s4[j] = make_float4(0.f, 0.f, 0.f, 0.f);
    for (int j = i; j < n_deg; j += stride) deg[j] = 0.f;
}

// ---------------------------------------------------------------------------
// Phase 1: deg[row] += edge_val[e]
// ---------------------------------------------------------------------------
__global__ void gcn_deg_kernel(const int* __restrict__ edge_row,
                               const float* __restrict__ edge_val,
                               float* __restrict__ deg, int E) {
    int e = blockIdx.x * blockDim.x + threadIdx.x;
    if (e < E) atomicAdd(&deg[edge_row[e]], edge_val[e]);
}

// ---------------------------------------------------------------------------
// Phase 2: scatter  support[row][:] += val * x[col][:]
// One wave32 per edge; each lane owns 4 contiguous features (float4 gather,
// 4 global_atomic_add_f32 that land in L2 since support fits in 192MB L2).
// ---------------------------------------------------------------------------
__global__ void gcn_scatter_kernel(const float* __restrict__ x,
                                   const int* __restrict__ edge_row,
                                   const int* __restrict__ edge_col,
                                   const float* __restrict__ edge_val,
                                   float* __restrict__ support, int E) {
    int tid  = blockIdx.x * blockDim.x + threadIdx.x;
    int e    = tid >> 5;          // wave32: one edge per wave
    int lane = tid & 31;
    if (e >= E) return;
    int   row = edge_row[e];
    int   col = edge_col[e];
    float val = edge_val[e];
    const float4* xs = (const float4*)(x + (size_t)col * F_IN);
    float4 v = xs[lane];
    float* dst = support + (size_t)row * F_IN + lane * 4;
    atomicAdd(dst + 0, val * v.x);
    atomicAdd(dst + 1, val * v.y);
    atomicAdd(dst + 2, val * v.z);
    atomicAdd(dst + 3, val * v.w);
}

// ---------------------------------------------------------------------------
// Phase 3: out = diag(deg^-1/2) * support @ W + bias  via V_WMMA_F32_16X16X4_F32
// Block = 256 threads = 8 waves. Block b owns rows [16b, 16b+16).
// Wave w computes the 16x16 output tile at columns [16w, 16w+16).
// support rows are staged (pre-scaled) into padded LDS; reads support from
// d_out and writes the matmul result back to the same rows (block-private).
// ---------------------------------------------------------------------------
#define TILE_LDS_PITCH 132   // 128 + 4 pad: A-fragment ds_load_b64 bank-conflict free

__global__ void gcn_matmul_kernel(const float* support,        // == out (aliased)
                                  const float* __restrict__ deg,
                                  const float* __restrict__ W, // [128][128] row-major
                                  const float* __restrict__ bias,
                                  float* out) {
    __shared__ float s_tile[16 * TILE_LDS_PITCH];

    const int m0 = blockIdx.x * 16;

    // --- stage 16x128 support rows into LDS, scaled by 1/sqrt(deg[row]) ---
    // 2048 elements / 256 threads = 8 per thread
    for (int j = 0; j < 8; ++j) {
        int idx = threadIdx.x + j * 256;
        int r = idx >> 7;            // 0..15
        int k = idx & 127;           // 0..127
        float dinv = __frsqrt_rn(deg[m0 + r]);
        s_tile[r * TILE_LDS_PITCH + k] = support[(size_t)(m0 + r) * F_IN + k] * dinv;
    }
    __syncthreads();

    const int lane = threadIdx.x & 31;
    const int wave = threadIdx.x >> 5;
    const int n0   = wave * 16;
    const int half = lane >> 4;      // 0: K pair {k0,k0+1}; 1: K pair {k0+2,k0+3}
    const int l    = lane & 15;      // M (A) / N (B,C) index within half-wave

    v8f c = {};
#pragma unroll
    for (int k0 = 0; k0 < F_IN; k0 += 4) {
        const int ka = k0 + half * 2;
        // A fragment: lane<16 -> row l, K={k0,k0+1}; lane>=16 -> K={k0+2,k0+3}
        v2f a = *(const v2f*)&s_tile[l * TILE_LDS_PITCH + ka];
        // B fragment: lane<16 -> rows {k0,k0+1} of W at col n0+l; lane>=16 -> {k0+2,k0+3}
        v2f b;
        b.x = W[(size_t)ka * F_OUT + n0 + l];
        b.y = W[(size_t)(ka + 1) * F_OUT + n0 + l];
        // D = A x B + C   (8 args: neg_a, A, neg_b, B, c_mod, C, reuse_a, reuse_b)
        c = __builtin_amdgcn_wmma_f32_16x16x4_f32(false, a, false, b, (short)0, c,
                                                  false, false);
    }

    // --- C/D layout: VGPR r -> row m0 + r + 8*half, col n0 + l ---
    const float bv = bias[n0 + l];
#pragma unroll
    for (int r = 0; r < 8; ++r) {
        int m = m0 + r + half * 8;
        out[(size_t)m * F_OUT + n0 + l] = c[r] + bv;
    }
}

// ---------------------------------------------------------------------------
extern "C" void kernel_launch(void* const* d_in, const int* in_sizes, int n_in,
                              void* d_out, int out_size, void* d_ws, size_t ws_size,
                              hipStream_t stream) {
    const float* x        = (const float*)d_in[0];
    const int*   edge_row = (const int*)  d_in[1];
    const int*   edge_col = (const int*)  d_in[2];
    const float* edge_val = (const float*)d_in[3];
    const float* weight   = (const float*)d_in[4];
    const float* bias     = (const float*)d_in[5];
    float* out = (float*)d_out;
    float* deg = (float*)d_ws;                 // N floats (400 KB) of scratch

    const int E = in_sizes[1];                 // number of edges
    const int N = out_size / F_OUT;            // number of nodes (100000)

    // Phase 0: zero support accumulator (d_out) + degree array
    gcn_zero_kernel<<<2048, 256, 0, stream>>>(out, deg, out_size, N);

    // Phase 1: degrees
    gcn_deg_kernel<<<(E + 255) / 256, 256, 0, stream>>>(edge_row, edge_val, deg, E);

    // Phase 2: SpMM scatter (one wave32 per edge)
    long long total = (long long)E * 32;
    int blocks = (int)((total + 255) / 256);
    gcn_scatter_kernel<<<blocks, 256, 0, stream>>>(x, edge_row, edge_col, edge_val, out, E);

    // Phase 3: row-scale + dense matmul + bias via fp32 WMMA (in-place on d_out)
    gcn_matmul_kernel<<<N / 16, 256, 0, stream>>>(out, deg, weight, bias, out);
}